// CustomGATLayer_25632364822806
// MI455X (gfx1250) — compile-verified
//
#include <hip/hip_runtime.h>
#include <math.h>

#define N_NODES 100000
#define N_EDGES 1600000
#define IN_DIM  128
#define HEADS   8
#define HDIM    16
#define OUT_DIM 128          /* HEADS * HDIM */
#define NEG_SLOPE 0.01f
#define LDS_STRIDE 130       /* pad so lane c hits bank 2c (mod 64): conflict-free ds_load_b64 */

typedef __attribute__((ext_vector_type(2))) float v2f;
typedef __attribute__((ext_vector_type(8))) float v8f;

// ---------------------------------------------------------------------------
// Float atomic max via signed/unsigned integer ordering trick.
// Correct for IEEE-754 when the cell is initialized to -INF.
// ---------------------------------------------------------------------------
__device__ __forceinline__ void atomicMaxF(float* addr, float val) {
    if (val >= 0.0f) {
        atomicMax((int*)addr, __float_as_int(val));
    } else {
        atomicMin((unsigned int*)addr, __float_as_uint(val));
    }
}

// ---------------------------------------------------------------------------
// Kernel 0: init per-node softmax state and zero the aggregation buffer (d_out)
// ---------------------------------------------------------------------------
__global__ void gat_init(float* __restrict__ out0, float* __restrict__ m,
                         float* __restrict__ den) {
    int j = blockIdx.x * blockDim.x + threadIdx.x;
    if (j < N_NODES * OUT_DIM) out0[j] = 0.0f;
    if (j < N_NODES * HEADS) { m[j] = -INFINITY; den[j] = 0.0f; }
}

// ---------------------------------------------------------------------------
// Kernel 1: z[n, h*16+d] = sum_i h[n,i] * Wfc[h,i,d]   via V_WMMA_F32_16X16X4_F32
// Block = 256 threads = 8 waves; W (128x128, repacked from Wfc) staged in LDS
// column-major. Each wave computes a 16x128 slab (8 WMMA accumulators).
// ---------------------------------------------------------------------------
__global__ void __launch_bounds__(256)
gat_gemm_wmma(const float* __restrict__ h, const float* __restrict__ Wfc,
              float* __restrict__ z) {
    __shared__ float Wl[OUT_DIM * LDS_STRIDE];   // Wl[col*130 + k]

    const int tid = threadIdx.x;
    // Repack Wfc[h][i][d] -> W[k=i][col=h*16+d], stored column-major in LDS.
    for (int j = tid; j < IN_DIM * OUT_DIM; j += 256) {
        int i  = j >> 7;          // k index 0..127
        int c  = j & 127;         // output column 0..127
        int hh = c >> 4;
        int d  = c & 15;
        Wl[c * LDS_STRIDE + i] = Wfc[hh * (IN_DIM * HDIM) + i * HDIM + d];
    }
    __syncthreads();

    const int wave  = tid >> 5;
    const int lane  = tid & 31;
    const int mtile = blockIdx.x * 8 + wave;
    if (mtile >= N_NODES / 16) return;          // wave-uniform: EXEC all-ones inside

    const int row   = lane & 15;                // M row for A, N col for B
    const int khalf = lane >> 4;                // 0: K={0,1}, 1: K={2,3}

    const float* hrow = h + (size_t)(mtile * 16 + row) * IN_DIM;

    v8f acc[8] = {};

    for (int k0 = 0; k0 < IN_DIM; k0 += 4) {
        const int kk = k0 + 2 * khalf;
        // A fragment: 16x4 f32, lane holds h[row][kk], h[row][kk+1]
        v2f a = *reinterpret_cast<const v2f*>(hrow + kk);
#pragma unroll
        for (int t = 0; t < 8; ++t) {
            // B fragment: 4x16 f32, lane holds W[kk][t*16+row], W[kk+1][t*16+row]
            v2f b = *reinterpret_cast<const v2f*>(&Wl[(t * 16 + row) * LDS_STRIDE + kk]);
            acc[t] = __builtin_amdgcn_wmma_f32_16x16x4_f32(
                false, a, false, b, (short)0, acc[t], false, false);
        }
    }

    // C/D layout: VGPR i, lanes 0-15 -> M=i, lanes 16-31 -> M=i+8, col = lane&15
    float* zbase = z + (size_t)(mtile * 16) * OUT_DIM;
#pragma unroll
    for (int t = 0; t < 8; ++t) {
#pragma unroll
        for (int i = 0; i < 8; ++i) {
            int mrow = i + khalf * 8;
            zbase[(size_t)mrow * OUT_DIM + t * 16 + row] = acc[t][i];
        }
    }
}

// ---------------------------------------------------------------------------
// Kernel 2: s_src[n,h] = z[n,h,:]·a_src[h],  s_dst[n,h] = z[n,h,:]·a_dst[h]
// ---------------------------------------------------------------------------
__global__ void gat_scores(const float* __restrict__ z,
                           const float* __restrict__ Wattn,
                           float* __restrict__ s_src, float* __restrict__ s_dst) {
    int idx = blockIdx.x * blockDim.x + threadIdx.x;
    if (idx >= N_NODES * HEADS) return;
    int n = idx >> 3, hh = idx & 7;
    const float* zp = z + (size_t)n * OUT_DIM + hh * HDIM;
    const float* as = Wattn + hh * (2 * HDIM);
    const float* ad = as + HDIM;
    float ss = 0.0f, sd = 0.0f;
#pragma unroll
    for (int d = 0; d < HDIM; ++d) {
        float v = zp[d];
        ss += v * as[d];
        sd += v * ad[d];
    }
    s_src[idx] = ss;
    s_dst[idx] = sd;
}

// ---------------------------------------------------------------------------
// Kernel 3: segment max of leaky-relu scores per (dst, head)
// ---------------------------------------------------------------------------
__global__ void gat_edge_max(const int* __restrict__ src, const int* __restrict__ dst,
                             const float* __restrict__ s_src, const float* __restrict__ s_dst,
                             float* __restrict__ m) {
    int e = blockIdx.x * blockDim.x + threadIdx.x;
    if (e >= N_EDGES) return;
    int s = src[e], d = dst[e];
    const float* ps = s_src + (size_t)s * HEADS;
    const float* pd = s_dst + (size_t)d * HEADS;
    float* pm = m + (size_t)d * HEADS;
#pragma unroll
    for (int hh = 0; hh < HEADS; ++hh) {
        float sc = ps[hh] + pd[hh];
        sc = (sc >= 0.0f) ? sc : NEG_SLOPE * sc;
        atomicMaxF(pm + hh, sc);
    }
}

// ---------------------------------------------------------------------------
// Kernel 4: den[d,h] += exp(sc - m[d,h])
// ---------------------------------------------------------------------------
__global__ void gat_edge_sum(const int* __restrict__ src, const int* __restrict__ dst,
                             const float* __restrict__ s_src, const float* __restrict__ s_dst,
                             const float* __restrict__ m, float* __restrict__ den) {
    int e = blockIdx.x * blockDim.x + threadIdx.x;
    if (e >= N_EDGES) return;
    int s = src[e], d = dst[e];
    const float* ps = s_src + (size_t)s * HEADS;
    const float* pd = s_dst + (size_t)d * HEADS;
    const float* pm = m + (size_t)d * HEADS;
    float* pden = den + (size_t)d * HEADS;
#pragma unroll
    for (int hh = 0; hh < HEADS; ++hh) {
        float sc = ps[hh] + pd[hh];
        sc = (sc >= 0.0f) ? sc : NEG_SLOPE * sc;
        atomicAdd(pden + hh, expf(sc - pm[hh]));
    }
}

// ---------------------------------------------------------------------------
// Kernel 5: out0[dst, h, :] += alpha * z[src, h, :]   (one thread per edge*head)
// ---------------------------------------------------------------------------
__global__ void gat_edge_agg(const int* __restrict__ src, const int* __restrict__ dst,
                             const float* __restrict__ s_src, const float* __restrict__ s_dst,
                             const float* __restrict__ m, const float* __restrict__ den,
                             const float* __restrict__ z, float* __restrict__ out0) {
    int idx = blockIdx.x * blockDim.x + threadIdx.x;
    if (idx >= N_EDGES * HEADS) return;
    int e = idx >> 3, hh = idx & 7;
    int s = src[e], d = dst[e];
    float sc = s_src[(size_t)s * HEADS + hh] + s_dst[(size_t)d * HEADS + hh];
    sc = (sc >= 0.0f) ? sc : NEG_SLOPE * sc;
    float alpha = expf(sc - m[(size_t)d * HEADS + hh]) / den[(size_t)d * HEADS + hh];

    const float4* zp = (const float4*)(z + (size_t)s * OUT_DIM + hh * HDIM);
    float* op = out0 + (size_t)d * OUT_DIM + hh * HDIM;
#pragma unroll
    for (int q = 0; q < 4; ++q) {
        float4 v = zp[q];
        atomicAdd(op + q * 4 + 0, alpha * v.x);
        atomicAdd(op + q * 4 + 1, alpha * v.y);
        atomicAdd(op + q * 4 + 2, alpha * v.z);
        atomicAdd(op + q * 4 + 3, alpha * v.w);
    }
}

// ---------------------------------------------------------------------------
// Kernel 6: h_out = h + elu(agg)   (in place in d_out), plus e pass-through
// ---------------------------------------------------------------------------
__global__ void gat_finalize(const float* __restrict__ h, const float* __restrict__ e,
                             float* __restrict__ out) {
    int j = blockIdx.x * blockDim.x + threadIdx.x;
    const int NH = N_NODES * OUT_DIM;
    if (j < NH) {
        float v = out[j];
        v = (v > 0.0f) ? v : expm1f(v);
        out[j] = h[j] + v;
    } else if (j < NH + N_EDGES) {
        out[j] = e[j - NH];
    }
}

// ---------------------------------------------------------------------------
extern "C" void kernel_launch(void* const* d_in, const int* in_sizes, int n_in,
                              void* d_out, int out_size, void* d_ws, size_t ws_size,
                              hipStream_t stream) {
    const float* h     = (const float*)d_in[0];
    const float* e     = (const float*)d_in[1];
    const int*   src   = (const int*)d_in[2];
    const int*   dst   = (const int*)d_in[3];
    const float* Wfc   = (const float*)d_in[4];
    const float* Wattn = (const float*)d_in[5];
    float* out = (float*)d_out;

    // Workspace layout (floats): z | s_src | s_dst | m | den  = 64 MB total
    float* z     = (float*)d_ws;
    float* s_src = z + (size_t)N_NODES * OUT_DIM;
    float* s_dst = s_src + (size_t)N_NODES * HEADS;
    float* m     = s_dst + (size_t)N_NODES * HEADS;
    float* den   = m + (size_t)N_NODES * HEADS;

    const int T = 256;

    gat_init<<<(N_NODES * OUT_DIM + T - 1) / T, T, 0, stream>>>(out, m, den);

    // 6250 M-tiles, 8 waves per block
    gat_gemm_wmma<<<(N_NODES / 16 + 7) / 8, T, 0, stream>>>(h, Wfc, z);

    gat_scores<<<(N_NODES * HEADS + T - 1) / T, T, 0, stream>>>(z, Wattn, s_src, s_dst);

    gat_edge_max<<<(N_EDGES + T - 1) / T, T, 0, stream>>>(src, dst, s_src, s_dst, m);
    gat_edge_sum<<<(N_EDGES + T - 1) / T, T, 0, stream>>>(src, dst, s_src, s_dst, m, den);
    gat_edge_agg<<<(N_EDGES * HEADS + T - 1) / T, T, 0, stream>>>(src, dst, s_src, s_dst,
                                                                  m, den, z, out);

    gat_finalize<<<(N_NODES * OUT_DIM + N_EDGES + T - 1) / T, T, 0, stream>>>(h, e, out);
}